// GNNModel_23733989278407
// MI455X (gfx1250) — compile-verified
//
#include <hip/hip_runtime.h>
#include <math.h>

typedef __attribute__((ext_vector_type(2))) float v2f;
typedef __attribute__((ext_vector_type(8))) float v8f;

#define HDIM 32
#define LN_EPS 1e-5f

// ---------------- degree / normalization ----------------
__global__ void k_deg_init(float* __restrict__ deg, int n) {
  int i = blockIdx.x * blockDim.x + threadIdx.x;
  if (i < n) deg[i] = 1.0f;  // self-loop contributes 1 to every node
}

__global__ void k_deg_count(const int* __restrict__ dst, float* __restrict__ deg, int e) {
  int stride = gridDim.x * blockDim.x;
  for (int i = blockIdx.x * blockDim.x + threadIdx.x; i < e; i += stride)
    atomicAdd(&deg[dst[i]], 1.0f);
}

__global__ void k_deg_rsqrt(float* __restrict__ deg, int n) {
  int i = blockIdx.x * blockDim.x + threadIdx.x;
  if (i < n) deg[i] = rsqrtf(deg[i]);  // deg >= 1 always (self-loops)
}

// ---------------- f32 WMMA GEMM: Y[nrows x 32] = X[nrows x K] * W[K x 32] ----
// One wave per 16x16 output tile; V_WMMA_F32_16X16X4_F32, loop K in steps of 4.
__global__ void k_gemm_wmma(const float* __restrict__ X, const float* __restrict__ W,
                            float* __restrict__ Y, int nrows, int K) {
  const int lane = threadIdx.x & 31;
  const int gw   = (blockIdx.x * blockDim.x + threadIdx.x) >> 5;
  const int tile_m = gw >> 1;        // 2 column tiles (HDIM = 32)
  const int tile_n = gw & 1;
  const int row0 = tile_m * 16;
  if (row0 >= nrows) return;         // wave-uniform exit: EXEC stays all-ones
  const int col0 = tile_n * 16;
  const int m  = lane & 15;          // row (A) / col (B) within tile
  const int kk = (lane >> 4) << 1;   // lanes 0-15 -> K=0,1 ; lanes 16-31 -> K=2,3

  int arow = row0 + m;
  if (arow >= nrows) arow = nrows - 1;   // clamp (no divergence)
  const float* xrow = X + (size_t)arow * K;

  v8f acc = {};
  for (int k0 = 0; k0 < K; k0 += 4) {
    v2f a, b;
    a.x = xrow[k0 + kk];
    a.y = xrow[k0 + kk + 1];
    b.x = W[(size_t)(k0 + kk)     * HDIM + col0 + m];
    b.y = W[(size_t)(k0 + kk + 1) * HDIM + col0 + m];
    acc = __builtin_amdgcn_wmma_f32_16x16x4_f32(
        /*neg_a=*/false, a, /*neg_b=*/false, b,
        /*c_mod=*/(short)0, acc, /*reuse_a=*/false, /*reuse_b=*/false);
  }

  // C/D layout: VGPR r -> row r (lanes 0-15) / row r+8 (lanes 16-31); col = lane&15
  const int rbase = (lane >> 4) << 3;
#pragma unroll
  for (int r = 0; r < 8; ++r) {
    int orow = row0 + rbase + r;
    if (orow < nrows) Y[(size_t)orow * HDIM + col0 + m] = acc[r];
  }
}

// ---------------- self-loop init: out = t * dis^2 ----------------
__global__ void k_self_init(const float* __restrict__ t, const float* __restrict__ dis,
                            float* __restrict__ out, int n) {
  int idx = blockIdx.x * blockDim.x + threadIdx.x;
  if (idx < n * HDIM) {
    float d = dis[idx >> 5];
    out[idx] = t[idx] * d * d;
  }
}

// ---------------- edge scatter-add: wave per edge, lane per feature ----------
__global__ void k_edge_agg(const int* __restrict__ src, const int* __restrict__ dst,
                           const float* __restrict__ dis, const float* __restrict__ h,
                           float* __restrict__ out, int e) {
  const int lane = threadIdx.x & 31;
  int gw = (blockIdx.x * blockDim.x + threadIdx.x) >> 5;
  int nw = (gridDim.x * blockDim.x) >> 5;
  for (int i = gw; i < e; i += nw) {
    int s = src[i];
    int d = dst[i];
    float nrm = dis[s] * dis[d];
    float v = h[(size_t)s * HDIM + lane] * nrm;
    atomicAdd(&out[(size_t)d * HDIM + lane], v);
  }
}

// ---------------- bias + ReLU + LayerNorm (wave per node, lane per feature) --
__global__ void k_epi_ln(const float* __restrict__ agg, const float* __restrict__ bias,
                         const float* __restrict__ g, const float* __restrict__ bv,
                         float* __restrict__ out, int n) {
  const int lane = threadIdx.x & 31;
  int node = (blockIdx.x * blockDim.x + threadIdx.x) >> 5;
  if (node >= n) return;
  float v = agg[(size_t)node * HDIM + lane] + bias[lane];
  v = fmaxf(v, 0.0f);
  float s = v;
#pragma unroll
  for (int off = 16; off > 0; off >>= 1) s += __shfl_xor(s, off, 32);
  float mean = s * (1.0f / 32.0f);
  float dv = v - mean;
  float sq = dv * dv;
#pragma unroll
  for (int off = 16; off > 0; off >>= 1) sq += __shfl_xor(sq, off, 32);
  float var = sq * (1.0f / 32.0f);
  float r = rsqrtf(var + LN_EPS);
  out[(size_t)node * HDIM + lane] = dv * r * g[lane] + bv[lane];
}

// ---------------- last layer bias+ReLU + post_mp head + log_softmax ---------
__global__ void k_head(const float* __restrict__ agg, const float* __restrict__ gb,
                       const float* __restrict__ w1, const float* __restrict__ b1,
                       const float* __restrict__ w2, const float* __restrict__ b2,
                       float* __restrict__ out, int n) {
  int node = blockIdx.x * blockDim.x + threadIdx.x;
  if (node >= n) return;
  const float* p = agg + (size_t)node * HDIM;
  float h[HDIM];
#pragma unroll
  for (int c = 0; c < HDIM; ++c) h[c] = fmaxf(p[c] + gb[c], 0.0f);

  float y1[16];
#pragma unroll
  for (int j = 0; j < 16; ++j) y1[j] = b1[j];
  for (int c = 0; c < HDIM; ++c) {
    float hc = h[c];
#pragma unroll
    for (int j = 0; j < 16; ++j) y1[j] = fmaf(hc, w1[c * 16 + j], y1[j]);
  }
#pragma unroll
  for (int j = 0; j < 16; ++j) y1[j] = fmaxf(y1[j], 0.0f);

  float y2[8];
#pragma unroll
  for (int j = 0; j < 8; ++j) y2[j] = b2[j];
  for (int c = 0; c < 16; ++c) {
    float yc = y1[c];
#pragma unroll
    for (int j = 0; j < 8; ++j) y2[j] = fmaf(yc, w2[c * 8 + j], y2[j]);
  }

  float mx = y2[0];
#pragma unroll
  for (int j = 1; j < 8; ++j) mx = fmaxf(mx, y2[j]);
  float se = 0.0f;
#pragma unroll
  for (int j = 0; j < 8; ++j) se += expf(y2[j] - mx);
  float lse = logf(se) + mx;
#pragma unroll
  for (int j = 0; j < 8; ++j) out[(size_t)node * 8 + j] = y2[j] - lse;
}

extern "C" void kernel_launch(void* const* d_in, const int* in_sizes, int n_in,
                              void* d_out, int out_size, void* d_ws, size_t ws_size,
                              hipStream_t stream) {
  const float* x     = (const float*)d_in[0];
  const int*   ei    = (const int*)d_in[1];
  const float* W0    = (const float*)d_in[2];
  const float* b0    = (const float*)d_in[3];
  const float* W1    = (const float*)d_in[4];
  const float* b1    = (const float*)d_in[5];
  const float* W2    = (const float*)d_in[6];
  const float* b2    = (const float*)d_in[7];
  const float* ln0g  = (const float*)d_in[8];
  const float* ln0b  = (const float*)d_in[9];
  const float* ln1g  = (const float*)d_in[10];
  const float* ln1b  = (const float*)d_in[11];
  const float* mpw1  = (const float*)d_in[12];
  const float* mpb1  = (const float*)d_in[13];
  const float* mpw2  = (const float*)d_in[14];
  const float* mpb2  = (const float*)d_in[15];

  const int DIN = in_sizes[2] / HDIM;       // 128
  const int N   = in_sizes[0] / DIN;        // 100000
  const int E   = in_sizes[1] / 2;          // 3200000
  const int* src = ei;
  const int* dst = ei + E;

  float* out = (float*)d_out;
  float* ws  = (float*)d_ws;
  float* dis = ws;                          // N
  float* A   = ws + N;                      // N*32
  float* B   = A + (size_t)N * HDIM;        // N*32

  const int node_blocks = (N + 255) / 256;
  const int elem_blocks = (N * HDIM + 255) / 256;   // wave-per-node launches too
  const int tiles       = ((N + 15) / 16) * 2;
  const int gemm_blocks = (tiles + 7) / 8;          // 8 waves / 256-thread block
  const int edge_blocks = 8192;

  // symmetric normalization: dis = rsqrt(deg + 1)
  k_deg_init <<<node_blocks, 256, 0, stream>>>(dis, N);
  k_deg_count<<<4096,        256, 0, stream>>>(dst, dis, E);
  k_deg_rsqrt<<<node_blocks, 256, 0, stream>>>(dis, N);

  // ---- layer 0 ----
  k_gemm_wmma<<<gemm_blocks, 256, 0, stream>>>(x, W0, A, N, DIN);
  k_self_init<<<elem_blocks, 256, 0, stream>>>(A, dis, B, N);
  k_edge_agg <<<edge_blocks, 256, 0, stream>>>(src, dst, dis, A, B, E);
  k_epi_ln   <<<elem_blocks, 256, 0, stream>>>(B, b0, ln0g, ln0b, A, N);

  // ---- layer 1 ----
  k_gemm_wmma<<<gemm_blocks, 256, 0, stream>>>(A, W1, B, N, HDIM);
  k_self_init<<<elem_blocks, 256, 0, stream>>>(B, dis, A, N);
  k_edge_agg <<<edge_blocks, 256, 0, stream>>>(src, dst, dis, B, A, E);
  k_epi_ln   <<<elem_blocks, 256, 0, stream>>>(A, b1, ln1g, ln1b, B, N);

  // ---- layer 2 (no LN) + head ----
  k_gemm_wmma<<<gemm_blocks, 256, 0, stream>>>(B, W2, A, N, HDIM);
  k_self_init<<<elem_blocks, 256, 0, stream>>>(A, dis, B, N);
  k_edge_agg <<<edge_blocks, 256, 0, stream>>>(src, dst, dis, A, B, E);
  k_head     <<<node_blocks, 256, 0, stream>>>(B, b2, mpw1, mpb1, mpw2, mpb2, out, N);
}